// PEGNNModel_34600256537257
// MI455X (gfx1250) — compile-verified
//
#include <hip/hip_runtime.h>
#include <hip/hip_bf16.h>

// ---------------------------------------------------------------------------
// GraphSAGE (3x SAGEConv mean-aggr + linear head) for MI455X / gfx1250.
// fp32 everywhere; GEMMs use V_WMMA_F32_16X16X4_F32 (exact fp32 matmul).
// ---------------------------------------------------------------------------

#define N_NODES 50000
#define N_EDGES 800000
#define CH      128
#define OUT_CH  16

typedef __attribute__((ext_vector_type(2))) float v2f;
typedef __attribute__((ext_vector_type(8))) float v8f;

// ---------------------------------------------------------------------------
// Degree: deg[dst] += 1 per edge (float so we can reciprocal directly).
// ---------------------------------------------------------------------------
__global__ __launch_bounds__(256) void degree_kernel(const int* __restrict__ dst,
                                                     float* __restrict__ deg,
                                                     int n_edges) {
    int e = blockIdx.x * blockDim.x + threadIdx.x;
    if (e < n_edges) atomicAdd(&deg[dst[e]], 1.0f);
}

// ---------------------------------------------------------------------------
// Scatter-add of source features into agg[dst]. 32 threads per edge, each
// thread moves 4 consecutive channels (float4 load + 4 f32 global atomics).
// Whole h / agg matrices (25.6 MB each) live in the 192 MB L2.
// ---------------------------------------------------------------------------
__global__ __launch_bounds__(256) void scatter_kernel(const float* __restrict__ h,
                                                      const int* __restrict__ src,
                                                      const int* __restrict__ dst,
                                                      float* __restrict__ agg,
                                                      int n_edges) {
    int t = blockIdx.x * blockDim.x + threadIdx.x;
    int e = t >> 5;
    if (e >= n_edges) return;
    int c = (t & 31) * 4;
    int s = src[e];
    int d = dst[e];
    const float4 v = *(const float4*)(h + (size_t)s * CH + c);
    float* p = agg + (size_t)d * CH + c;
    atomicAdd(p + 0, v.x);
    atomicAdd(p + 1, v.y);
    atomicAdd(p + 2, v.z);
    atomicAdd(p + 3, v.w);
}

// ---------------------------------------------------------------------------
// SAGE layer GEMM: out = relu( (agg * 1/max(deg,1)) @ Wl + xin @ Wr + bias )
// One wave per 16x16 output tile; 8 waves/block -> one 16-row node tile's
// full 16x128 output. K=128 in steps of 4 via V_WMMA_F32_16X16X4_F32.
//
// A layout (16x4 f32): lanes 0-15 row M=l16, v0/v1 = K=k0,k0+1;
//                      lanes 16-31 row M=l16, v0/v1 = K=k0+2,k0+3.
// B layout (4x16 f32): N striped across lanes, same K split by lane half.
// C/D: VGPR r = row (r + 8*half), col = n0 + l16.
// ---------------------------------------------------------------------------
__global__ __launch_bounds__(256) void sage_gemm_kernel(const float* __restrict__ agg,
                                                        const float* __restrict__ deg,
                                                        const float* __restrict__ xin,
                                                        const float* __restrict__ Wl,
                                                        const float* __restrict__ Wr,
                                                        const float* __restrict__ bias,
                                                        float* __restrict__ out) {
    const int wave = threadIdx.x >> 5;       // 0..7  -> N tile
    const int lane = threadIdx.x & 31;
    const int half = lane >> 4;              // 0: lanes 0-15, 1: lanes 16-31
    const int l16  = lane & 15;
    const int row0 = blockIdx.x * 16;        // node tile base
    const int n0   = wave * 16;              // output-column tile base

    const int arow = row0 + l16;             // A row this lane feeds
    const float invd = 1.0f / fmaxf(deg[arow], 1.0f);

    const float* aggr = agg + (size_t)arow * CH;
    const float* xr   = xin + (size_t)arow * CH;

    v8f acc = {};
    #pragma unroll 4
    for (int k0 = 0; k0 < CH; k0 += 4) {
        const int ka = k0 + half * 2;
        v2f a, b;
        // neighbor-mean term
        a.x = aggr[ka]     * invd;
        a.y = aggr[ka + 1] * invd;
        b.x = Wl[(size_t)ka       * CH + n0 + l16];
        b.y = Wl[(size_t)(ka + 1) * CH + n0 + l16];
        acc = __builtin_amdgcn_wmma_f32_16x16x4_f32(false, a, false, b,
                                                    (short)0, acc, false, false);
        // root term
        a.x = xr[ka];
        a.y = xr[ka + 1];
        b.x = Wr[(size_t)ka       * CH + n0 + l16];
        b.y = Wr[(size_t)(ka + 1) * CH + n0 + l16];
        acc = __builtin_amdgcn_wmma_f32_16x16x4_f32(false, a, false, b,
                                                    (short)0, acc, false, false);
    }

    const float bv = bias[n0 + l16];
    #pragma unroll
    for (int r = 0; r < 8; ++r) {
        float v = acc[r] + bv;
        v = fmaxf(v, 0.0f);                  // relu
        out[(size_t)(row0 + r + 8 * half) * CH + n0 + l16] = v;
    }
}

// ---------------------------------------------------------------------------
// Head: out[N,16] = h @ W_lin(128x16) + b_lin. One wave per 16-node tile,
// single 16-wide N tile. Early-exit is wave-uniform (EXEC stays all-ones).
// ---------------------------------------------------------------------------
__global__ __launch_bounds__(256) void head_gemm_kernel(const float* __restrict__ h,
                                                        const float* __restrict__ W,
                                                        const float* __restrict__ bias,
                                                        float* __restrict__ out,
                                                        int n_nodes) {
    const int wave = threadIdx.x >> 5;
    const int lane = threadIdx.x & 31;
    const int half = lane >> 4;
    const int l16  = lane & 15;
    const int tile = blockIdx.x * 8 + wave;
    const int row0 = tile * 16;
    if (row0 >= n_nodes) return;             // uniform per wave

    const float* hr = h + (size_t)(row0 + l16) * CH;

    v8f acc = {};
    #pragma unroll 4
    for (int k0 = 0; k0 < CH; k0 += 4) {
        const int ka = k0 + half * 2;
        v2f a, b;
        a.x = hr[ka];
        a.y = hr[ka + 1];
        b.x = W[(size_t)ka       * OUT_CH + l16];
        b.y = W[(size_t)(ka + 1) * OUT_CH + l16];
        acc = __builtin_amdgcn_wmma_f32_16x16x4_f32(false, a, false, b,
                                                    (short)0, acc, false, false);
    }

    const float bv = bias[l16];
    #pragma unroll
    for (int r = 0; r < 8; ++r) {
        out[(size_t)(row0 + r + 8 * half) * OUT_CH + l16] = acc[r] + bv;
    }
}

// ---------------------------------------------------------------------------
// Host-side orchestration (graph-capture safe: only async ops on `stream`).
// ---------------------------------------------------------------------------
extern "C" void kernel_launch(void* const* d_in, const int* in_sizes, int n_in,
                              void* d_out, int out_size, void* d_ws, size_t ws_size,
                              hipStream_t stream) {
    const float* x    = (const float*)d_in[0];
    const int*   ei   = (const int*)d_in[1];       // [2, N_EDGES]
    const int*   src  = ei;
    const int*   dst  = ei + N_EDGES;
    const float* W1l  = (const float*)d_in[2];
    const float* b1   = (const float*)d_in[3];
    const float* W1r  = (const float*)d_in[4];
    const float* W2l  = (const float*)d_in[5];
    const float* b2   = (const float*)d_in[6];
    const float* W2r  = (const float*)d_in[7];
    const float* W3l  = (const float*)d_in[8];
    const float* b3   = (const float*)d_in[9];
    const float* W3r  = (const float*)d_in[10];
    const float* Wlin = (const float*)d_in[11];
    const float* blin = (const float*)d_in[12];

    // Workspace layout (all fp32): deg[N] | agg[N*128] | h1[N*128] | h2[N*128]
    float* deg = (float*)d_ws;
    float* agg = deg + N_NODES;
    float* h1  = agg + (size_t)N_NODES * CH;
    float* h2  = h1  + (size_t)N_NODES * CH;

    const size_t featBytes = (size_t)N_NODES * CH * sizeof(float);

    const int tiles       = N_NODES / 16;                  // 3125 exactly
    const int degBlocks   = (N_EDGES + 255) / 256;
    const int scatBlocks  = (int)(((size_t)N_EDGES * 32 + 255) / 256);
    const int headBlocks  = (tiles + 7) / 8;

    hipMemsetAsync(deg, 0, N_NODES * sizeof(float), stream);
    hipMemsetAsync(agg, 0, featBytes, stream);
    degree_kernel<<<degBlocks, 256, 0, stream>>>(dst, deg, N_EDGES);

    // ---- Layer 1: x -> h1
    scatter_kernel<<<scatBlocks, 256, 0, stream>>>(x, src, dst, agg, N_EDGES);
    sage_gemm_kernel<<<tiles, 256, 0, stream>>>(agg, deg, x, W1l, W1r, b1, h1);

    // ---- Layer 2: h1 -> h2
    hipMemsetAsync(agg, 0, featBytes, stream);
    scatter_kernel<<<scatBlocks, 256, 0, stream>>>(h1, src, dst, agg, N_EDGES);
    sage_gemm_kernel<<<tiles, 256, 0, stream>>>(agg, deg, h1, W2l, W2r, b2, h2);

    // ---- Layer 3: h2 -> h1 (reuse)
    hipMemsetAsync(agg, 0, featBytes, stream);
    scatter_kernel<<<scatBlocks, 256, 0, stream>>>(h2, src, dst, agg, N_EDGES);
    sage_gemm_kernel<<<tiles, 256, 0, stream>>>(agg, deg, h2, W3l, W3r, b3, h1);

    // ---- Head: h1 @ W_lin + b_lin -> d_out [N,16]
    head_gemm_kernel<<<headBlocks, 256, 0, stream>>>(h1, Wlin, blin, (float*)d_out, N_NODES);
}